// PopulationEncoder_44092134260938
// MI455X (gfx1250) — compile-verified
//
#include <hip/hip_runtime.h>
#include <hip/hip_bf16.h>
#include <math.h>
#include <stdint.h>

#define TIME_STEPS   100
#define NUM_NEURONS  100
#define BF           8192            // 64 * 128
#define TOTAL_ROWS   (BF * NUM_NEURONS)          // 819,200 (b,f,n) rows
#define QUADS_PER_ROW 25                         // 100 timesteps / 4
#define TOTAL_THREADS 20480000u                  // TOTAL_ROWS * 25

typedef __attribute__((ext_vector_type(2))) float v2f;
typedef __attribute__((ext_vector_type(4))) float v4f;
typedef __attribute__((ext_vector_type(8))) float v8f;

// ---------------------------------------------------------------------------
// Philox 4x32-10: counter-based, deterministic per flat element index.
// ---------------------------------------------------------------------------
__device__ __forceinline__ void philox_round(uint32_t& c0, uint32_t& c1,
                                             uint32_t& c2, uint32_t& c3,
                                             uint32_t k0, uint32_t k1) {
  const uint32_t M0 = 0xD2511F53u, M1 = 0xCD9E8D57u;
  uint32_t lo0 = M0 * c0;
  uint32_t hi0 = __umulhi(M0, c0);
  uint32_t lo1 = M1 * c2;
  uint32_t hi1 = __umulhi(M1, c2);
  c0 = hi1 ^ c1 ^ k0;
  c1 = lo1;
  c2 = hi0 ^ c3 ^ k1;
  c3 = lo0;
}

__device__ __forceinline__ void philox10(uint32_t gid, uint32_t& c0, uint32_t& c1,
                                         uint32_t& c2, uint32_t& c3) {
  c0 = gid; c1 = 0x853C49E6u; c2 = 0xDA3E39CBu; c3 = 0xB5297A4Du;
  uint32_t k0 = 0xA511E9B3u, k1 = 0x8B5F7C39u;
#pragma unroll
  for (int r = 0; r < 10; ++r) {
    philox_round(c0, c1, c2, c3, k0, k1);
    k0 += 0x9E3779B9u;   // golden-ratio Weyl
    k1 += 0xBB67AE85u;
  }
}

// ---------------------------------------------------------------------------
// Kernel 1: rates[bf, n] = exp(-(x-c)^2 / (2 sigma^2)) * max_rate / 1000
// Squared distance via one V_WMMA_F32_16X16X4_F32 per 16x16 tile:
//   (x - c)^2 = x^2*1 + x*(-2c) + 1*c^2      (K = 3 of 4 used)
// A 16x4 layout: lane L holds row M=L&15; VGPR0/1 = K={0,1} (L<16) or {2,3}.
// B 4x16 layout: lane L holds col N=L&15; VGPR0/1 = K={0,1} (L<16) or {2,3}.
// D layout: VGPR v in lane L is element (M = v + 8*(L>=16), N = L&15).
// One wave per tile; EXEC all-ones as WMMA requires (bounds handled at store).
// ---------------------------------------------------------------------------
__global__ __launch_bounds__(32) void pe_rates_wmma(
    const float* __restrict__ x, const float* __restrict__ centers,
    const float* __restrict__ log_sigma, const float* __restrict__ log_max_rate,
    float* __restrict__ rates) {
  const int lane    = threadIdx.x;          // 0..31, wave32
  const int half    = lane >> 4;            // 0: K=0,1   1: K=2,3
  const int idx     = lane & 15;
  const int rowBase = blockIdx.x * 16;      // over BF = 8192 (exact multiple)
  const int colBase = blockIdx.y * 16;      // over N (padded to 112)

  const float xv = x[rowBase + idx];
  const int   n  = colBase + idx;
  const float cv = centers[n < NUM_NEURONS ? n : NUM_NEURONS - 1];

  v2f a, b;
  if (half == 0) {                          // K=0: x^2 * 1 ; K=1: x * (-2c)
    a.x = xv * xv;  a.y = xv;
    b.x = 1.0f;     b.y = -2.0f * cv;
  } else {                                  // K=2: 1 * c^2 ; K=3: unused
    a.x = 1.0f;     a.y = 0.0f;
    b.x = cv * cv;  b.y = 0.0f;
  }

  v8f d = {};
  d = __builtin_amdgcn_wmma_f32_16x16x4_f32(
      /*neg_a=*/false, a, /*neg_b=*/false, b,
      /*c_mod=*/(short)0, d, /*reuse_a=*/false, /*reuse_b=*/false);

  const float sigma = expf(log_sigma[0]);
  const float alpha = -1.0f / (2.0f * sigma * sigma);
  const float scale = expf(log_max_rate[0]) * 1.0e-3f;

  if (n < NUM_NEURONS) {
#pragma unroll
    for (int v = 0; v < 8; ++v) {
      const int row = rowBase + v + half * 8;
      rates[(size_t)row * NUM_NEURONS + n] = scale * expf(alpha * d[v]);
    }
  }
}

// ---------------------------------------------------------------------------
// Kernel 2: spikes. One thread = 4 timesteps of one (b,f,n) row.
// 327.7 MB output > 192 MB L2 -> nontemporal 16B stores.
// ---------------------------------------------------------------------------
__global__ __launch_bounds__(256) void pe_spikes(
    const float* __restrict__ rates, float* __restrict__ out) {
  const uint32_t gid = blockIdx.x * 256u + threadIdx.x;
  if (gid >= TOTAL_THREADS) return;
  const uint32_t row = gid / QUADS_PER_ROW;      // (b,f,n) flat index
  const uint32_t q   = gid - row * QUADS_PER_ROW;

  const float rate = rates[row];                 // L2-resident, broadcast

  uint32_t c0, c1, c2, c3;
  philox10(gid, c0, c1, c2, c3);

  const float inv24 = 1.0f / 16777216.0f;        // top 24 bits -> [0,1)
  v4f s;
  s.x = ((float)(c0 >> 8) * inv24 < rate) ? 1.0f : 0.0f;
  s.y = ((float)(c1 >> 8) * inv24 < rate) ? 1.0f : 0.0f;
  s.z = ((float)(c2 >> 8) * inv24 < rate) ? 1.0f : 0.0f;
  s.w = ((float)(c3 >> 8) * inv24 < rate) ? 1.0f : 0.0f;

  v4f* p = (v4f*)(out + (size_t)row * TIME_STEPS + (size_t)q * 4);
  __builtin_nontemporal_store(s, p);             // global_store_b128, TH=NT
}

// ---------------------------------------------------------------------------
// Fallback: fused single-pass (used only if workspace is too small).
// ---------------------------------------------------------------------------
__global__ __launch_bounds__(256) void pe_spikes_fused(
    const float* __restrict__ x, const float* __restrict__ centers,
    const float* __restrict__ log_sigma, const float* __restrict__ log_max_rate,
    float* __restrict__ out) {
  const uint32_t gid = blockIdx.x * 256u + threadIdx.x;
  if (gid >= TOTAL_THREADS) return;
  const uint32_t row = gid / QUADS_PER_ROW;
  const uint32_t q   = gid - row * QUADS_PER_ROW;
  const uint32_t bf  = row / NUM_NEURONS;
  const uint32_t n   = row - bf * NUM_NEURONS;

  const float sigma = expf(log_sigma[0]);
  const float alpha = -1.0f / (2.0f * sigma * sigma);
  const float scale = expf(log_max_rate[0]) * 1.0e-3f;
  const float diff  = x[bf] - centers[n];
  const float rate  = scale * expf(alpha * diff * diff);

  uint32_t c0, c1, c2, c3;
  philox10(gid, c0, c1, c2, c3);

  const float inv24 = 1.0f / 16777216.0f;
  v4f s;
  s.x = ((float)(c0 >> 8) * inv24 < rate) ? 1.0f : 0.0f;
  s.y = ((float)(c1 >> 8) * inv24 < rate) ? 1.0f : 0.0f;
  s.z = ((float)(c2 >> 8) * inv24 < rate) ? 1.0f : 0.0f;
  s.w = ((float)(c3 >> 8) * inv24 < rate) ? 1.0f : 0.0f;

  v4f* p = (v4f*)(out + (size_t)row * TIME_STEPS + (size_t)q * 4);
  __builtin_nontemporal_store(s, p);
}

// ---------------------------------------------------------------------------
extern "C" void kernel_launch(void* const* d_in, const int* in_sizes, int n_in,
                              void* d_out, int out_size, void* d_ws, size_t ws_size,
                              hipStream_t stream) {
  const float* x            = (const float*)d_in[0];   // [64,128]
  const float* centers      = (const float*)d_in[1];   // [100]
  const float* log_sigma    = (const float*)d_in[2];   // scalar
  const float* log_max_rate = (const float*)d_in[3];   // scalar
  float*       out          = (float*)d_out;           // [64,128,100,100]

  const size_t rates_bytes = (size_t)TOTAL_ROWS * sizeof(float);  // 3.2 MB
  const unsigned blocks = (TOTAL_THREADS + 255u) / 256u;          // 80,000

  if (d_ws != nullptr && ws_size >= rates_bytes) {
    float* rates = (float*)d_ws;
    dim3 g1(BF / 16, (NUM_NEURONS + 15) / 16);        // 512 x 7 tiles, 1 wave each
    pe_rates_wmma<<<g1, 32, 0, stream>>>(x, centers, log_sigma, log_max_rate, rates);
    pe_spikes<<<blocks, 256, 0, stream>>>(rates, out);
  } else {
    pe_spikes_fused<<<blocks, 256, 0, stream>>>(x, centers, log_sigma, log_max_rate, out);
  }
}